// LightGCN_47725676593618
// MI455X (gfx1250) — compile-verified
//
#include <hip/hip_runtime.h>

typedef __attribute__((ext_vector_type(2))) float v2f;
typedef __attribute__((ext_vector_type(8))) float v8f;

#define DIM 64

// ---------------------------------------------------------------- utilities
__global__ void zero_kernel(float* __restrict__ p, long long n) {
  long long i = (long long)blockIdx.x * blockDim.x + threadIdx.x;
  if (i < n) p[i] = 0.0f;
}

// ------------------------------------------------------------ degree (atomic)
__global__ void deg_kernel(const int* __restrict__ eu, const int* __restrict__ ei,
                           const float* __restrict__ ev, float* __restrict__ deg,
                           int U, int E) {
  int e = blockIdx.x * blockDim.x + threadIdx.x;
  if (e >= E) return;
  float v = ev[e];
  atomicAdd(&deg[eu[e]], v);
  atomicAdd(&deg[U + ei[e]], v);
}

__global__ void dinv_kernel(const float* __restrict__ deg, float* __restrict__ dinv, int N) {
  int i = blockIdx.x * blockDim.x + threadIdx.x;
  if (i >= N) return;
  float d = deg[i];
  dinv[i] = (d > 0.0f) ? (1.0f / sqrtf(fmaxf(d, 1e-12f))) : 0.0f;
}

// -------------------------------------------- symmetric edge coefficient c[e]
__global__ void coeff_kernel(const int* __restrict__ eu, const int* __restrict__ ei,
                             const float* __restrict__ ev, const float* __restrict__ dinv,
                             float* __restrict__ c, int U, int E) {
  int e = blockIdx.x * blockDim.x + threadIdx.x;
  if (e >= E) return;
  c[e] = dinv[eu[e]] * ev[e] * dinv[U + ei[e]];
}

// ---- init: e_cur = emb ; acc slots of d_out = emb ; verbatim emb copies ----
// d_out layout: [embf_users U*64][emb_users U*64][embf_items I*64][emb_items I*64]
__global__ void init_kernel(const float* __restrict__ emb_u, const float* __restrict__ emb_i,
                            float* __restrict__ ecur, float* __restrict__ out, int U, int I) {
  long long idx = (long long)blockIdx.x * blockDim.x + threadIdx.x;
  long long total = (long long)(U + I) * DIM;
  if (idx >= total) return;
  int n = (int)(idx >> 6);
  int d = (int)(idx & 63);
  float v = (n < U) ? emb_u[(long long)n * DIM + d]
                    : emb_i[(long long)(n - U) * DIM + d];
  ecur[idx] = v;
  long long accOff = (long long)((n < U) ? n : n + U) * DIM + d;        // embf slot (acc)
  out[accOff] = v;
  long long cpyOff = (long long)((n < U) ? (U + n) : (U + I + n)) * DIM + d; // emb copy
  out[cpyOff] = v;
}

// -------- one propagation layer: e_next[row] += c * e_cur[col], both dirs ----
// one wave32 per edge, float2 (2 dims) per lane; edge indices are wave-uniform
__global__ void scatter_kernel(const int* __restrict__ eu, const int* __restrict__ ei,
                               const float* __restrict__ c, const float* __restrict__ ecur,
                               float* __restrict__ enext, int U, int E) {
  long long t = (long long)blockIdx.x * blockDim.x + threadIdx.x;
  int e = (int)(t >> 5);
  int lane = (int)(t & 31);
  if (e >= E) return;
  if (lane == 0 && (e + 4096) < E) {           // stream-ahead on edge arrays
    __builtin_prefetch(eu + e + 4096, 0, 1);   // -> global_prefetch_b8
    __builtin_prefetch(ei + e + 4096, 0, 1);
    __builtin_prefetch(c + e + 4096, 0, 1);
  }
  int a = eu[e];
  int b = U + ei[e];
  float w = c[e];
  const float2 va = *((const float2*)(ecur + (long long)a * DIM) + lane);
  const float2 vb = *((const float2*)(ecur + (long long)b * DIM) + lane);
  float* qa = enext + (long long)a * DIM + 2 * lane;
  float* qb = enext + (long long)b * DIM + 2 * lane;
  atomicAdd(qa + 0, w * vb.x);
  atomicAdd(qa + 1, w * vb.y);
  atomicAdd(qb + 0, w * va.x);
  atomicAdd(qb + 1, w * va.y);
}

// ---------------------------- acc += e_next (acc lives in d_out embf slots) --
__global__ void accadd_kernel(float* __restrict__ out, const float* __restrict__ enext,
                              int U, long long total) {
  long long idx = (long long)blockIdx.x * blockDim.x + threadIdx.x;
  if (idx >= total) return;
  int n = (int)(idx >> 6);
  int d = (int)(idx & 63);
  long long o = (long long)((n < U) ? n : n + U) * DIM + d;
  out[o] += enext[idx];
}

// ------------- final scale by s via V_WMMA_F32_16X16X4_F32 (exact f32) -------
// Per wave: one 16(node)x16(dim) tile T.  D = sum_j A_j*B_j with
// A_j[m,k] = s * (m == 4j+k), B_j[k,n] = T[4j+k, n]  =>  D = s*T.
// Layouts per ISA 7.12.2: A lane=M (both halves), VGPR0/1 = K{0,1} lo-half /
// K{2,3} hi-half; B lane=N, same K split; C/D VGPR r -> row r (lo) / r+8 (hi).
__global__ void scale_wmma_kernel(float* __restrict__ out, int U, int nTiles, float s) {
  int wave = (int)(((long long)blockIdx.x * blockDim.x + threadIdx.x) >> 5);
  int lane = (int)(threadIdx.x & 31);
  if (wave >= nTiles) return;               // wave-uniform: EXEC stays all-ones
  int trow = wave >> 2;                     // DIM/16 == 4 tiles per node-row
  int tcol = wave & 3;
  int node0 = trow * 16;
  long long obase = (long long)((node0 < U) ? node0 : node0 + U) * DIM + tcol * 16;
  float* T = out + obase;
  int half = lane >> 4;                     // 0: K=0,1   1: K=2,3
  int l15 = lane & 15;                      // A's M and B's N
  v8f acc = {};
#pragma unroll
  for (int j = 0; j < 4; ++j) {
    int r0 = 4 * j + half * 2;              // first tile-row this half handles
    v2f a, b;
    a.x = (l15 == r0)     ? s : 0.0f;
    a.y = (l15 == r0 + 1) ? s : 0.0f;
    b.x = T[(long long)(r0)     * DIM + l15];
    b.y = T[(long long)(r0 + 1) * DIM + l15];
    acc = __builtin_amdgcn_wmma_f32_16x16x4_f32(false, a, false, b,
                                                (short)0, acc, false, false);
  }
#pragma unroll
  for (int r = 0; r < 8; ++r) {
    T[(long long)(r + half * 8) * DIM + l15] = acc[r];
  }
}

// scalar fallback / tail (only used if 16 does not divide U or N)
__global__ void scale_tail_kernel(float* __restrict__ out, int U, int nodeStart,
                                  int N, float s) {
  long long idx = (long long)blockIdx.x * blockDim.x + threadIdx.x;
  long long total = (long long)(N - nodeStart) * DIM;
  if (idx >= total) return;
  int n = nodeStart + (int)(idx >> 6);
  int d = (int)(idx & 63);
  long long o = (long long)((n < U) ? n : n + U) * DIM + d;
  out[o] *= s;
}

// ---------------------------------------------------------------------- host
extern "C" void kernel_launch(void* const* d_in, const int* in_sizes, int n_in,
                              void* d_out, int out_size, void* d_ws, size_t ws_size,
                              hipStream_t stream) {
  (void)n_in; (void)out_size; (void)ws_size;
  const float* emb_u = (const float*)d_in[0];
  const float* emb_i = (const float*)d_in[1];
  const int*   eu    = (const int*)d_in[2];
  const int*   ei    = (const int*)d_in[3];
  const float* ev    = (const float*)d_in[4];

  const int U = in_sizes[0] / DIM;
  const int I = in_sizes[1] / DIM;
  const int E = in_sizes[2];
  const int N = U + I;
  const long long N64 = (long long)N * DIM;

  float* ws   = (float*)d_ws;
  float* deg  = ws;                 // N
  float* dinv = deg + N;            // N
  float* c    = dinv + N;           // E
  float* buf0 = c + E;              // N*64
  float* buf1 = buf0 + N64;         // N*64
  float* out  = (float*)d_out;

  const int B = 256;                // 8 wave32s per workgroup
  auto blk = [](long long n, int b) { return (int)((n + b - 1) / b); };

  zero_kernel <<<blk(N, B),   B, 0, stream>>>(deg, N);
  deg_kernel  <<<blk(E, B),   B, 0, stream>>>(eu, ei, ev, deg, U, E);
  dinv_kernel <<<blk(N, B),   B, 0, stream>>>(deg, dinv, N);
  coeff_kernel<<<blk(E, B),   B, 0, stream>>>(eu, ei, ev, dinv, c, U, E);
  init_kernel <<<blk(N64, B), B, 0, stream>>>(emb_u, emb_i, buf0, out, U, I);

  float* cur = buf0;
  float* nxt = buf1;
  for (int layer = 0; layer < 4; ++layer) {
    zero_kernel   <<<blk(N64, B), B, 0, stream>>>(nxt, N64);
    scatter_kernel<<<blk((long long)E * 32, B), B, 0, stream>>>(eu, ei, c, cur, nxt, U, E);
    accadd_kernel <<<blk(N64, B), B, 0, stream>>>(out, nxt, U, N64);
    float* t = cur; cur = nxt; nxt = t;
  }

  const float s = 1.0f / 25.0f;     // alpha * (1/(L+1)) with L=4
  if ((U % 16 == 0) && (N % 16 == 0)) {
    int nTiles = (N / 16) * (DIM / 16);
    scale_wmma_kernel<<<blk((long long)nTiles * 32, B), B, 0, stream>>>(out, U, nTiles, s);
  } else {
    scale_tail_kernel<<<blk(N64, B), B, 0, stream>>>(out, U, 0, N, s);
  }
}